// GraphNet1_16080357556242
// MI455X (gfx1250) — compile-verified
//
#include <hip/hip_runtime.h>
#include <cstddef>

#define NN   100000
#define NE   1600000
#define CIN  128
#define HIDD 64
#define EPSf 1e-5f

typedef __attribute__((ext_vector_type(2))) float v2f;
typedef __attribute__((ext_vector_type(8))) float v8f;

// ---- workspace layout (float offsets) ----
#define OFF_H     ((size_t)0)                    // NN*HIDD
#define OFF_OUT   ((size_t)NN * HIDD)            // NN*HIDD
#define OFF_DEG   ((size_t)2 * NN * HIDD)        // NN (deg, then dinv in place)
#define OFF_SUM0  (OFF_DEG + NN)                 // CIN
#define OFF_SSQ0  (OFF_SUM0 + CIN)               // CIN
#define OFF_S0    (OFF_SSQ0 + CIN)               // CIN
#define OFF_T0    (OFF_S0 + CIN)                 // CIN
#define OFF_WPF   (OFF_T0 + CIN)                 // CIN*HIDD (fragment-ordered W')
#define OFF_TW    (OFF_WPF + CIN * HIDD)         // HIDD
#define OFF_SUM1  (OFF_TW + HIDD)                // HIDD
#define OFF_SSQ1  (OFF_SUM1 + HIDD)              // HIDD
#define OFF_S1    (OFF_SSQ1 + HIDD)              // HIDD
#define OFF_T1    (OFF_S1 + HIDD)                // HIDD

// 1) per-call re-init: deg=1 (self loop), zero stat accumulators
__global__ void k_init(float* deg, float* sum0, float* sum1) {
    int i = blockIdx.x * blockDim.x + threadIdx.x;
    if (i < NN) deg[i] = 1.0f;
    if (i < 2 * CIN)  sum0[i] = 0.0f;   // sum0 + ssq0 contiguous
    if (i < 2 * HIDD) sum1[i] = 0.0f;   // sum1 + ssq1 contiguous
}

// 2) BN0 stats over x[NN][CIN]
__global__ void k_bn0_stats(const float* __restrict__ x, float* sum, float* ssq) {
    int c = threadIdx.x;                       // 0..127
    int stride = gridDim.x * blockDim.y;
    float a = 0.f, a2 = 0.f;
    for (int r = blockIdx.x * blockDim.y + threadIdx.y; r < NN; r += stride) {
        float v = x[(size_t)r * CIN + c];
        a += v; a2 += v * v;
    }
    atomicAdd(&sum[c], a);
    atomicAdd(&ssq[c], a2);
}

// 3) degree histogram over dst (col)
__global__ void k_deg(const int* __restrict__ ei, float* deg) {
    int e = blockIdx.x * blockDim.x + threadIdx.x;
    if (e < NE) atomicAdd(&deg[ei[NE + e]], 1.0f);
}

// 4/11) BN affine finalize: s = gamma*rsqrt(var+eps), t = beta - mean*s
__global__ void k_bn_finalize(const float* sum, const float* ssq,
                              const float* gamma, const float* beta,
                              float* s, float* t) {
    int c = threadIdx.x;
    float mean = sum[c] * (1.0f / NN);
    float var  = ssq[c] * (1.0f / NN) - mean * mean;
    float sc   = gamma[c] * rsqrtf(var + EPSf);
    s[c] = sc;
    t[c] = beta[c] - mean * sc;
}

// 5) fold BN0 into weights, emitted directly in WMMA B-fragment order:
//    WpF[(kb*4 + nt)*32 + lane] (v2f) = { s0[c0]*W[c0][j], s0[c0+1]*W[c0+1][j] }
//    where kk = (lane<16)?0:2, c0 = 4*kb+kk, j = 16*nt + (lane&15).
//    Also tW[j] = sum_c t0[c]*W[c][j].
__global__ void k_prep_w(const float* __restrict__ W, const float* s0, const float* t0,
                         float* WpF, float* tW) {
    int tid = threadIdx.x;                     // 256 threads
    for (int idx = tid; idx < (CIN * HIDD) / 2; idx += 256) {   // 4096 fragments
        int lane = idx & 31;
        int nt   = (idx >> 5) & 3;
        int kb   = idx >> 7;
        int kk   = (lane < 16) ? 0 : 2;
        int j    = nt * 16 + (lane & 15);
        int c0   = 4 * kb + kk;
        WpF[2 * idx + 0] = s0[c0]     * W[c0 * HIDD + j];
        WpF[2 * idx + 1] = s0[c0 + 1] * W[(c0 + 1) * HIDD + j];
    }
    if (tid < HIDD) {
        float acc = 0.f;
        for (int c = 0; c < CIN; ++c) acc += t0[c] * W[c * HIDD + tid];
        tW[tid] = acc;
    }
}

// 6) deg -> dinv in place (deg >= 1 always due to self loop)
__global__ void k_dinv(float* deg) {
    int i = blockIdx.x * blockDim.x + threadIdx.x;
    if (i < NN) deg[i] = rsqrtf(deg[i]);
}

// 7) h = x @ Wp + tW via V_WMMA_F32_16X16X4_F32.
//    blockDim = 128 (4 waves); each wave computes one 16x64 output tile.
//    W' staged into LDS with GLOBAL_LOAD_ASYNC_TO_LDS_B128 (ASYNCcnt path);
//    each B fragment is then a single conflict-free ds_load_b64.
__global__ void k_gemm_wmma(const float* __restrict__ x,
                            const float* __restrict__ WpF,
                            const float* __restrict__ tW,
                            float* __restrict__ h) {
    __shared__ float sB[CIN * HIDD];           // 32 KB, fragment order
    const int tid = threadIdx.x;

    // async stage: 2048 x 16B, 128 threads -> 16 issues/lane, coalesced
    {
        unsigned ldsbase = (unsigned)(size_t)(&sB[0]);   // low 32b = LDS offset
        #pragma unroll
        for (int j = 0; j < 16; ++j) {
            int i4 = tid + 128 * j;
            unsigned loff = ldsbase + i4 * 16;
            unsigned long long ga = (unsigned long long)(size_t)(WpF + 4 * i4);
            asm volatile("global_load_async_to_lds_b128 %0, %1, off"
                         :: "v"(loff), "v"(ga) : "memory");
        }
        asm volatile("s_wait_asynccnt 0x0" ::: "memory");
    }
    __syncthreads();

    const int wave = tid >> 5;
    const int lane = tid & 31;
    const int lrow = lane & 15;
    const int row0 = blockIdx.x * 64 + wave * 16;

    // A-fragment (16x4 f32): lanes 0-15 hold K=0,1; lanes 16-31 hold K=2,3
    int grow = row0 + lrow; if (grow > NN - 1) grow = NN - 1;   // clamp: EXEC all-1s
    const int kk = (lane < 16) ? 0 : 2;
    const float* xr = x + (size_t)grow * CIN + kk;

    float tw0 = tW[ 0 + lrow], tw1 = tW[16 + lrow],
          tw2 = tW[32 + lrow], tw3 = tW[48 + lrow];

    const v2f* sBv = (const v2f*)sB;
    v8f acc0 = {}, acc1 = {}, acc2 = {}, acc3 = {};
    #pragma unroll 4
    for (int kb = 0; kb < 32; ++kb) {
        v2f a;
        a.x = xr[4 * kb + 0];
        a.y = xr[4 * kb + 1];
        const int fb = kb * 128 + lane;        // v2f index: [kb][nt][lane]
        v2f b0 = sBv[fb +  0];
        v2f b1 = sBv[fb + 32];
        v2f b2 = sBv[fb + 64];
        v2f b3 = sBv[fb + 96];
        acc0 = __builtin_amdgcn_wmma_f32_16x16x4_f32(false, a, false, b0, (short)0, acc0, false, false);
        acc1 = __builtin_amdgcn_wmma_f32_16x16x4_f32(false, a, false, b1, (short)0, acc1, false, false);
        acc2 = __builtin_amdgcn_wmma_f32_16x16x4_f32(false, a, false, b2, (short)0, acc2, false, false);
        acc3 = __builtin_amdgcn_wmma_f32_16x16x4_f32(false, a, false, b3, (short)0, acc3, false, false);
    }

    // C/D 16x16 f32 layout: VGPR r -> M = r (lanes 0-15) / r+8 (lanes 16-31), N = lane&15
    const int rbase = row0 + ((lane < 16) ? 0 : 8);
    #pragma unroll
    for (int r = 0; r < 8; ++r) {
        const int rr = rbase + r;
        if (rr < NN) {
            float* hrow = h + (size_t)rr * HIDD + lrow;
            hrow[ 0] = acc0[r] + tw0;
            hrow[16] = acc1[r] + tw1;
            hrow[32] = acc2[r] + tw2;
            hrow[48] = acc3[r] + tw3;
        }
    }
}

// 8) out = b + h * dinv^2  (bias + self-loop message)
__global__ void k_init_out(const float* __restrict__ h, const float* __restrict__ dinv,
                           const float* __restrict__ b, float* __restrict__ aggr) {
    int i = blockIdx.x * 256 + threadIdx.x;
    if (i < NN * HIDD) {
        int node = i >> 6, c = i & 63;
        float d = dinv[node];
        aggr[i] = b[c] + h[i] * d * d;
    }
}

// 9) edge scatter: aggr[col] += h[row] * dinv[row]*dinv[col]
__global__ void k_edge_agg(const int* __restrict__ ei, const float* __restrict__ h,
                           const float* __restrict__ dinv, float* __restrict__ aggr) {
    int t = blockIdx.x * 256 + threadIdx.x;          // up to 102.4M < 2^31
    if (t < NE * HIDD) {
        int e = t >> 6, c = t & 63;
        int r  = ei[e];
        int cl = ei[NE + e];
        float nrm = dinv[r] * dinv[cl];
        atomicAdd(&aggr[(size_t)cl * HIDD + c], h[(size_t)r * HIDD + c] * nrm);
    }
}

// 10) ReLU in place + BN1 stats
__global__ void k_bn1_stats(float* __restrict__ aggr, float* sum, float* ssq) {
    int c = threadIdx.x;                              // 0..63
    int stride = gridDim.x * blockDim.y;
    float a = 0.f, a2 = 0.f;
    for (int r = blockIdx.x * blockDim.y + threadIdx.y; r < NN; r += stride) {
        size_t idx = (size_t)r * HIDD + c;
        float v = aggr[idx];
        v = v > 0.f ? v : 0.f;
        aggr[idx] = v;
        a += v; a2 += v * v;
    }
    atomicAdd(&sum[c], a);
    atomicAdd(&ssq[c], a2);
}

// 12) final BN1 apply -> d_out
__global__ void k_bn1_apply(const float* __restrict__ aggr, const float* s, const float* t,
                            float* __restrict__ out) {
    int i = blockIdx.x * 256 + threadIdx.x;
    if (i < NN * HIDD) {
        int c = i & 63;
        out[i] = aggr[i] * s[c] + t[c];
    }
}

extern "C" void kernel_launch(void* const* d_in, const int* in_sizes, int n_in,
                              void* d_out, int out_size, void* d_ws, size_t ws_size,
                              hipStream_t stream) {
    const float* x      = (const float*)d_in[0];
    const int*   ei     = (const int*)  d_in[1];
    const float* gamma0 = (const float*)d_in[2];
    const float* beta0  = (const float*)d_in[3];
    const float* W      = (const float*)d_in[4];
    const float* bvec   = (const float*)d_in[5];
    const float* gamma1 = (const float*)d_in[6];
    const float* beta1  = (const float*)d_in[7];
    float* out = (float*)d_out;
    float* ws  = (float*)d_ws;

    float* h    = ws + OFF_H;
    float* aggr = ws + OFF_OUT;
    float* deg  = ws + OFF_DEG;   // becomes dinv
    float* sum0 = ws + OFF_SUM0;
    float* ssq0 = ws + OFF_SSQ0;
    float* s0   = ws + OFF_S0;
    float* t0   = ws + OFF_T0;
    float* WpF  = ws + OFF_WPF;
    float* tW   = ws + OFF_TW;
    float* sum1 = ws + OFF_SUM1;
    float* ssq1 = ws + OFF_SSQ1;
    float* s1   = ws + OFF_S1;
    float* t1   = ws + OFF_T1;

    k_init<<<(NN + 255) / 256, 256, 0, stream>>>(deg, sum0, sum1);
    k_bn0_stats<<<256, dim3(128, 2), 0, stream>>>(x, sum0, ssq0);
    k_deg<<<(NE + 255) / 256, 256, 0, stream>>>(ei, deg);
    k_bn_finalize<<<1, CIN, 0, stream>>>(sum0, ssq0, gamma0, beta0, s0, t0);
    k_prep_w<<<1, 256, 0, stream>>>(W, s0, t0, WpF, tW);
    k_dinv<<<(NN + 255) / 256, 256, 0, stream>>>(deg);
    k_gemm_wmma<<<(NN + 63) / 64, 128, 0, stream>>>(x, WpF, tW, h);
    k_init_out<<<(NN * HIDD + 255) / 256, 256, 0, stream>>>(h, deg, bvec, aggr);
    k_edge_agg<<<(NE * HIDD + 255) / 256, 256, 0, stream>>>(ei, h, deg, aggr);
    k_bn1_stats<<<256, dim3(64, 4), 0, stream>>>(aggr, sum1, ssq1);
    k_bn_finalize<<<1, HIDD, 0, stream>>>(sum1, ssq1, gamma1, beta1, s1, t1);
    k_bn1_apply<<<(NN * HIDD + 255) / 256, 256, 0, stream>>>(aggr, s1, t1, out);
}